// edge_cond_conv_24953759990467
// MI455X (gfx1250) — compile-verified
//
#include <hip/hip_runtime.h>

#define N_NODES 10000
#define N_EDGES 30000
#define DIN     64
#define EDIM    16
#define H1      128
#define H2      64
#define C_OUT   10
#define C_PAD   16

typedef __attribute__((ext_vector_type(2))) float v2f;
typedef __attribute__((ext_vector_type(8))) float v8f;

// D = A(16x4 f32) * B(4x16 f32) + C(16x16 f32), wave32.
__device__ __forceinline__ v8f wmma4(v2f a, v2f b, v8f c) {
  return __builtin_amdgcn_wmma_f32_16x16x4_f32(
      /*neg_a=*/false, a, /*neg_b=*/false, b,
      /*c_mod=*/(short)0, c, /*reuse_a=*/false, /*reuse_b=*/false);
}

// ---------------------------------------------------------------------------
// K0: relayout We1 [k][i*H2+o] -> We1T [(i*16+k)][o]          (131072 floats)
//     relayout We2 [k][i*C+o]  -> We2T [(i*16+k)][o'] padded  (16384 floats)
// ---------------------------------------------------------------------------
__global__ void k_relayout(const float* __restrict__ We1,
                           const float* __restrict__ We2,
                           float* __restrict__ We1T,
                           float* __restrict__ We2T) {
  int idx = blockIdx.x * blockDim.x + threadIdx.x;
  const int n1 = (H1 * EDIM) * H2;   // 131072
  const int n2 = (H2 * EDIM) * C_PAD; // 16384
  if (idx < n1) {
    int K = idx / H2, o = idx % H2;
    int k = K & 15, i = K >> 4;
    We1T[idx] = We1[k * (H1 * H2) + i * H2 + o];
  } else if (idx < n1 + n2) {
    int j = idx - n1;
    int K = j >> 4, o = j & 15;
    int k = K & 15, i = K >> 4;
    We2T[j] = (o < C_OUT) ? We2[k * (H2 * C_OUT) + i * C_OUT + o] : 0.0f;
  }
}

// ---------------------------------------------------------------------------
// K1: h1 = relu(x @ W1 + b1), [10000,64]x[64,128]. 1 wave per 16x16 tile.
// grid = 625 (m-tiles), block = 256 = 8 waves (one per n-tile).
// ---------------------------------------------------------------------------
__global__ void k_h1(const float* __restrict__ x, const float* __restrict__ W1,
                     const float* __restrict__ b1, float* __restrict__ h1) {
  int lane = threadIdx.x & 31;
  int wave = threadIdx.x >> 5;      // n-tile 0..7
  int mt   = blockIdx.x;            // m-tile 0..624
  int l    = lane & 15;
  int koff = (lane < 16) ? 0 : 2;
  int cb   = wave * 16;
  const float* Arow = x + (size_t)(mt * 16 + l) * DIN;
  v8f acc = {};
  for (int k0 = 0; k0 < DIN; k0 += 4) {
    v2f a; a.x = Arow[k0 + koff];               a.y = Arow[k0 + koff + 1];
    v2f b; b.x = W1[(k0 + koff) * H1 + cb + l]; b.y = W1[(k0 + koff + 1) * H1 + cb + l];
    acc = wmma4(a, b, acc);
  }
  int col = cb + l;
  float bias = b1[col];
  int rbase = mt * 16 + ((lane < 16) ? 0 : 8);
  for (int r = 0; r < 8; ++r) {
    float v = acc[r] + bias;
    h1[(size_t)(rbase + r) * H1 + col] = v > 0.0f ? v : 0.0f;
  }
}

// ---------------------------------------------------------------------------
// K2: fused NNConv1 edge kernel. 1 wave per 16-edge tile.
// msg[16,64] = A[16,2048] @ We1T[2048,64] + hs[16,128] @ reshape(be1)[128,64]
// with A[m, i*16+k] = hs[m][i] * es[m][k]  (built on the fly from LDS).
// Scatter: atomic f32 adds into agg1[dst].
// ---------------------------------------------------------------------------
__global__ void k_conv1(const float* __restrict__ h1, const float* __restrict__ ea,
                        const int* __restrict__ src, const int* __restrict__ dst,
                        const float* __restrict__ We1T, const float* __restrict__ be1,
                        float* __restrict__ agg1) {
  __shared__ float hs[16][H1];
  __shared__ float es[16][EDIM];
  int lane  = threadIdx.x;          // blockDim = 32
  int ebase = blockIdx.x * 16;
  for (int i = lane; i < 16 * EDIM; i += 32)
    es[i >> 4][i & 15] = ea[(size_t)ebase * EDIM + i];
  for (int i = lane; i < 16 * H1; i += 32) {
    int e = i >> 7, col = i & 127;
    hs[e][col] = h1[(size_t)src[ebase + e] * H1 + col];
  }
  __syncthreads();

  int l    = lane & 15;
  int koff = (lane < 16) ? 0 : 2;
  v8f acc[4] = {};
  // main contraction over K = EDIM*H1 = 2048
  for (int k0 = 0; k0 < H1 * EDIM; k0 += 4) {
    int Ka = k0 + koff, Kb = Ka + 1;
    v2f a;
    a.x = hs[l][Ka >> 4] * es[l][Ka & 15];
    a.y = hs[l][Kb >> 4] * es[l][Kb & 15];
    const float* B0 = We1T + (size_t)Ka * H2;
    const float* B1 = We1T + (size_t)Kb * H2;
    for (int t = 0; t < 4; ++t) {
      v2f b; b.x = B0[t * 16 + l]; b.y = B1[t * 16 + l];
      acc[t] = wmma4(a, b, acc[t]);
    }
  }
  // edge-MLP bias contribution: hs @ Be1 (Be1[i][o] = be1[i*H2+o]), K = H1
  for (int k0 = 0; k0 < H1; k0 += 4) {
    int Ka = k0 + koff, Kb = Ka + 1;
    v2f a; a.x = hs[l][Ka]; a.y = hs[l][Kb];
    for (int t = 0; t < 4; ++t) {
      v2f b; b.x = be1[Ka * H2 + t * 16 + l]; b.y = be1[Kb * H2 + t * 16 + l];
      acc[t] = wmma4(a, b, acc[t]);
    }
  }
  int rbase = (lane < 16) ? 0 : 8;
  for (int t = 0; t < 4; ++t) {
    int col = t * 16 + l;
    for (int r = 0; r < 8; ++r) {
      int e = ebase + rbase + r;
      atomicAdd(&agg1[(size_t)dst[e] * H2 + col], acc[t][r]);
    }
  }
}

// ---------------------------------------------------------------------------
// K3: h2 = relu(agg1 + h1 @ root1 + bias1). grid=625, block=128 (4 n-tiles).
// ---------------------------------------------------------------------------
__global__ void k_h2(const float* __restrict__ h1, const float* __restrict__ root1,
                     const float* __restrict__ bias1, const float* __restrict__ agg1,
                     float* __restrict__ h2) {
  int lane = threadIdx.x & 31;
  int wave = threadIdx.x >> 5;      // n-tile 0..3
  int mt   = blockIdx.x;
  int l    = lane & 15;
  int koff = (lane < 16) ? 0 : 2;
  int cb   = wave * 16;
  const float* Arow = h1 + (size_t)(mt * 16 + l) * H1;
  v8f acc = {};
  for (int k0 = 0; k0 < H1; k0 += 4) {
    v2f a; a.x = Arow[k0 + koff];                  a.y = Arow[k0 + koff + 1];
    v2f b; b.x = root1[(k0 + koff) * H2 + cb + l]; b.y = root1[(k0 + koff + 1) * H2 + cb + l];
    acc = wmma4(a, b, acc);
  }
  int col = cb + l;
  float bias = bias1[col];
  int rbase = mt * 16 + ((lane < 16) ? 0 : 8);
  for (int r = 0; r < 8; ++r) {
    int row = rbase + r;
    float v = acc[r] + bias + agg1[(size_t)row * H2 + col];
    h2[(size_t)row * H2 + col] = v > 0.0f ? v : 0.0f;
  }
}

// ---------------------------------------------------------------------------
// K4: fused NNConv2 edge kernel. K = EDIM*H2 = 1024, output cols padded to 16.
// ---------------------------------------------------------------------------
__global__ void k_conv2(const float* __restrict__ h2, const float* __restrict__ ea,
                        const int* __restrict__ src, const int* __restrict__ dst,
                        const float* __restrict__ We2T, const float* __restrict__ be2,
                        float* __restrict__ agg2) {
  __shared__ float hs[16][H2];
  __shared__ float es[16][EDIM];
  int lane  = threadIdx.x;          // blockDim = 32
  int ebase = blockIdx.x * 16;
  for (int i = lane; i < 16 * EDIM; i += 32)
    es[i >> 4][i & 15] = ea[(size_t)ebase * EDIM + i];
  for (int i = lane; i < 16 * H2; i += 32) {
    int e = i >> 6, col = i & 63;
    hs[e][col] = h2[(size_t)src[ebase + e] * H2 + col];
  }
  __syncthreads();

  int l    = lane & 15;
  int koff = (lane < 16) ? 0 : 2;
  v8f acc = {};
  for (int k0 = 0; k0 < H2 * EDIM; k0 += 4) {
    int Ka = k0 + koff, Kb = Ka + 1;
    v2f a;
    a.x = hs[l][Ka >> 4] * es[l][Ka & 15];
    a.y = hs[l][Kb >> 4] * es[l][Kb & 15];
    v2f b; b.x = We2T[Ka * C_PAD + l]; b.y = We2T[Kb * C_PAD + l];
    acc = wmma4(a, b, acc);
  }
  // edge-MLP bias: hs @ Be2 (Be2[i][o] = be2[i*C_OUT+o]), padded columns = 0
  for (int k0 = 0; k0 < H2; k0 += 4) {
    int Ka = k0 + koff, Kb = Ka + 1;
    v2f a; a.x = hs[l][Ka]; a.y = hs[l][Kb];
    v2f b;
    b.x = (l < C_OUT) ? be2[Ka * C_OUT + l] : 0.0f;
    b.y = (l < C_OUT) ? be2[Kb * C_OUT + l] : 0.0f;
    acc = wmma4(a, b, acc);
  }
  int rbase = (lane < 16) ? 0 : 8;
  for (int r = 0; r < 8; ++r) {
    int e = ebase + rbase + r;
    atomicAdd(&agg2[(size_t)dst[e] * C_PAD + l], acc[r]);  // padded cols harmless
  }
}

// ---------------------------------------------------------------------------
// K5: z = agg2 + h2 @ root2 + bias2; out = log_softmax(z). 1 thread / node.
// ---------------------------------------------------------------------------
__global__ void k_final(const float* __restrict__ h2, const float* __restrict__ root2,
                        const float* __restrict__ bias2, const float* __restrict__ agg2,
                        float* __restrict__ out) {
  int n = blockIdx.x * blockDim.x + threadIdx.x;
  if (n >= N_NODES) return;
  float z[C_OUT];
  for (int o = 0; o < C_OUT; ++o) z[o] = agg2[(size_t)n * C_PAD + o] + bias2[o];
  const float* hrow = h2 + (size_t)n * H2;
  for (int i = 0; i < H2; ++i) {
    float hv = hrow[i];
    for (int o = 0; o < C_OUT; ++o) z[o] = fmaf(hv, root2[i * C_OUT + o], z[o]);
  }
  float m = z[0];
  for (int o = 1; o < C_OUT; ++o) m = fmaxf(m, z[o]);
  float s = 0.0f;
  for (int o = 0; o < C_OUT; ++o) s += __expf(z[o] - m);
  float lse = __logf(s) + m;
  for (int o = 0; o < C_OUT; ++o) out[(size_t)n * C_OUT + o] = z[o] - lse;
}

// ---------------------------------------------------------------------------
extern "C" void kernel_launch(void* const* d_in, const int* in_sizes, int n_in,
                              void* d_out, int out_size, void* d_ws, size_t ws_size,
                              hipStream_t stream) {
  const float* x     = (const float*)d_in[0];
  const float* ea    = (const float*)d_in[1];
  const int*   ei    = (const int*)  d_in[2];
  const float* W1    = (const float*)d_in[3];
  const float* b1    = (const float*)d_in[4];
  const float* We1   = (const float*)d_in[5];
  const float* be1   = (const float*)d_in[6];
  const float* root1 = (const float*)d_in[7];
  const float* bias1 = (const float*)d_in[8];
  const float* We2   = (const float*)d_in[9];
  const float* be2   = (const float*)d_in[10];
  const float* root2 = (const float*)d_in[11];
  const float* bias2 = (const float*)d_in[12];
  float* out = (float*)d_out;

  float* ws   = (float*)d_ws;
  size_t off  = 0;
  float* We1T = ws + off; off += (size_t)(H1 * EDIM) * H2;     // 131072
  float* We2T = ws + off; off += (size_t)(H2 * EDIM) * C_PAD;  // 16384
  float* h1v  = ws + off; off += (size_t)N_NODES * H1;         // 1.28M
  float* agg1 = ws + off; off += (size_t)N_NODES * H2;         // 640K
  float* h2v  = ws + off; off += (size_t)N_NODES * H2;         // 640K
  float* agg2 = ws + off; off += (size_t)N_NODES * C_PAD;      // 160K

  const int* src = ei;
  const int* dst = ei + N_EDGES;

  hipMemsetAsync(agg1, 0, (size_t)N_NODES * H2 * sizeof(float), stream);
  hipMemsetAsync(agg2, 0, (size_t)N_NODES * C_PAD * sizeof(float), stream);

  int relayout_total = (H1 * EDIM) * H2 + (H2 * EDIM) * C_PAD;
  k_relayout<<<(relayout_total + 255) / 256, 256, 0, stream>>>(We1, We2, We1T, We2T);
  k_h1   <<<N_NODES / 16, 256, 0, stream>>>(x, W1, b1, h1v);
  k_conv1<<<N_EDGES / 16,  32, 0, stream>>>(h1v, ea, src, dst, We1T, be1, agg1);
  k_h2   <<<N_NODES / 16, 128, 0, stream>>>(h1v, root1, bias1, agg1, h2v);
  k_conv2<<<N_EDGES / 16,  32, 0, stream>>>(h2v, ea, src, dst, We2T, be2, agg2);
  k_final<<<(N_NODES + 255) / 256, 256, 0, stream>>>(h2v, root2, bias2, agg2, out);
}